// GCNT0_76046690942999
// MI455X (gfx1250) — compile-verified
//
#include <hip/hip_runtime.h>

typedef __attribute__((ext_vector_type(2))) float v2f;
typedef __attribute__((ext_vector_type(8))) float v8f;

// ---------------- degree / normalization ----------------

__global__ __launch_bounds__(256) void k_deg_init(float* __restrict__ deg, int n) {
  int i = blockIdx.x * 256 + threadIdx.x;
  if (i < n) deg[i] = 1.0f;  // self loop
}

__global__ __launch_bounds__(256) void k_deg_count(const int* __restrict__ src,
                                                   const int* __restrict__ dst,
                                                   float* __restrict__ deg, int E) {
  int e = blockIdx.x * 256 + threadIdx.x;
  if (e < E) {
    atomicAdd(&deg[src[e]], 1.0f);
    atomicAdd(&deg[dst[e]], 1.0f);
  }
}

__global__ __launch_bounds__(256) void k_rsqrt(float* __restrict__ deg, int n) {
  int i = blockIdx.x * 256 + threadIdx.x;
  if (i < n) deg[i] = rsqrtf(deg[i]);
}

// ---------------- GEMM1: [n,128] @ [128,64] + b, fp32 WMMA ----------------
// Block: 256 threads = 8 waves; 32 rows per block; wave (mt,nt) owns a 16x16 tile.

__global__ __launch_bounds__(256) void k_gemm1(const float* __restrict__ X,
                                               const float* __restrict__ W,
                                               const float* __restrict__ b,
                                               float* __restrict__ H, int n) {
  constexpr int K = 128, NC = 64;
  __shared__ float Ws[K][NC + 1];
  __shared__ float Xs[32][K + 4];
  int tid = threadIdx.x;
  for (int i = tid; i < K * NC; i += 256) Ws[i / NC][i % NC] = W[i];
  int row0 = blockIdx.x * 32;
  for (int i = tid; i < 32 * K; i += 256) {
    int r = i / K, c = i % K, g = row0 + r;
    Xs[r][c] = (g < n) ? X[(size_t)g * K + c] : 0.0f;
  }
  __syncthreads();

  int lane = tid & 31, wave = tid >> 5;
  int mt = wave >> 2, nt = wave & 3;       // 2 m-tiles x 4 n-tiles
  int l16 = lane & 15, kg = lane >> 4;     // A/B fragment: K pair = kg*2
  v8f acc = {};
  for (int k0 = 0; k0 < K; k0 += 4) {
    v2f a, bb;
    a.x = Xs[mt * 16 + l16][k0 + kg * 2 + 0];
    a.y = Xs[mt * 16 + l16][k0 + kg * 2 + 1];
    bb.x = Ws[k0 + kg * 2 + 0][nt * 16 + l16];
    bb.y = Ws[k0 + kg * 2 + 1][nt * 16 + l16];
    acc = __builtin_amdgcn_wmma_f32_16x16x4_f32(false, a, false, bb,
                                                (short)0, acc, false, false);
  }
  int ncol = nt * 16 + l16;
  float bias = b[ncol];
#pragma unroll
  for (int r = 0; r < 8; ++r) {
    int row = row0 + mt * 16 + r + 8 * kg;  // C/D: VGPR r -> M=r (lanes 0-15), r+8 (16-31)
    if (row < n) H[(size_t)row * NC + ncol] = acc[r] + bias;
  }
}

// ---------------- GEMM2: [n,64] @ [64,40] + b, fp32 WMMA (N padded to 48) ----------------
// Block: 192 threads = 6 waves; 32 rows per block; 2 m-tiles x 3 n-tiles.

__global__ __launch_bounds__(192) void k_gemm2(const float* __restrict__ A,
                                               const float* __restrict__ W,
                                               const float* __restrict__ b,
                                               float* __restrict__ H, int n) {
  constexpr int K = 64, NOUT = 40, NPAD = 48;
  __shared__ float Ws[K][NPAD + 1];
  __shared__ float As[32][K + 4];
  int tid = threadIdx.x;
  for (int i = tid; i < K * NPAD; i += 192) {
    int r = i / NPAD, c = i % NPAD;
    Ws[r][c] = (c < NOUT) ? W[r * NOUT + c] : 0.0f;
  }
  int row0 = blockIdx.x * 32;
  for (int i = tid; i < 32 * K; i += 192) {
    int r = i / K, c = i % K, g = row0 + r;
    As[r][c] = (g < n) ? A[(size_t)g * K + c] : 0.0f;
  }
  __syncthreads();

  int lane = tid & 31, wave = tid >> 5;
  int mt = wave / 3, nt = wave % 3;
  int l16 = lane & 15, kg = lane >> 4;
  v8f acc = {};
  for (int k0 = 0; k0 < K; k0 += 4) {
    v2f a, bb;
    a.x = As[mt * 16 + l16][k0 + kg * 2 + 0];
    a.y = As[mt * 16 + l16][k0 + kg * 2 + 1];
    bb.x = Ws[k0 + kg * 2 + 0][nt * 16 + l16];
    bb.y = Ws[k0 + kg * 2 + 1][nt * 16 + l16];
    acc = __builtin_amdgcn_wmma_f32_16x16x4_f32(false, a, false, bb,
                                                (short)0, acc, false, false);
  }
  int ncol = nt * 16 + l16;
  if (ncol < NOUT) {            // divergence only at store; WMMA ran with full EXEC
    float bias = b[ncol];
#pragma unroll
    for (int r = 0; r < 8; ++r) {
      int row = row0 + mt * 16 + r + 8 * kg;
      if (row < n) H[(size_t)row * NOUT + ncol] = acc[r] + bias;
    }
  }
}

// ---------------- smoothing: self-loop init + edge scatter ----------------

__global__ __launch_bounds__(256) void k_smooth_init(const float* __restrict__ h,
                                                     const float* __restrict__ dis,
                                                     float* __restrict__ out, int n, int C) {
  int total = n * C;
  for (int i = blockIdx.x * 256 + threadIdx.x; i < total; i += gridDim.x * 256) {
    float s = dis[i / C];
    out[i] = h[i] * s * s;
  }
}

__global__ __launch_bounds__(256) void k_edge_scatter(const float* __restrict__ h,
                                                      const float* __restrict__ dis,
                                                      const int* __restrict__ src,
                                                      const int* __restrict__ dst,
                                                      float* __restrict__ out, int E, int C) {
  int total = E * C;
  for (int i = blockIdx.x * 256 + threadIdx.x; i < total; i += gridDim.x * 256) {
    int e = i / C, c = i - e * C;
    int s = src[e], d = dst[e];
    float norm = dis[s] * dis[d];
    atomicAdd(&out[(size_t)d * C + c], h[(size_t)s * C + c] * norm);
    atomicAdd(&out[(size_t)s * C + c], h[(size_t)d * C + c] * norm);
  }
}

__global__ __launch_bounds__(256) void k_relu(const float* __restrict__ in,
                                              float* __restrict__ out, int total) {
  int i = blockIdx.x * 256 + threadIdx.x;
  if (i < total) out[i] = fmaxf(in[i], 0.0f);
}

// ---------------- launch ----------------

extern "C" void kernel_launch(void* const* d_in, const int* in_sizes, int n_in,
                              void* d_out, int out_size, void* d_ws, size_t ws_size,
                              hipStream_t stream) {
  (void)n_in; (void)out_size; (void)ws_size;
  const float* X  = (const float*)d_in[0];
  const float* W1 = (const float*)d_in[1];
  const float* b1 = (const float*)d_in[2];
  const float* W2 = (const float*)d_in[3];
  const float* b2 = (const float*)d_in[4];
  const int*   ei = (const int*)d_in[5];

  const int Chid = in_sizes[2];            // 64
  const int Cin  = in_sizes[1] / Chid;     // 128
  const int Cout = in_sizes[4];            // 40
  const int N    = in_sizes[0] / Cin;      // 50000
  const int E    = in_sizes[5] / 2;        // 800000
  const int* src = ei;
  const int* dst = ei + E;

  float* dis  = (float*)d_ws;                  // N   (deg, then d^-1/2 in place)
  float* bufA = dis + N;                       // N*Chid (h1, later h2)
  float* bufB = bufA + (size_t)N * Chid;       // N*Chid (smoothed pre-ReLU)
  float* out1 = (float*)d_out;                 // N*Chid
  float* out2 = out1 + (size_t)N * Chid;       // N*Cout

  // normalization
  k_deg_init<<<(N + 255) / 256, 256, 0, stream>>>(dis, N);
  k_deg_count<<<(E + 255) / 256, 256, 0, stream>>>(src, dst, dis, E);
  k_rsqrt<<<(N + 255) / 256, 256, 0, stream>>>(dis, N);

  // layer 1
  k_gemm1<<<(N + 31) / 32, 256, 0, stream>>>(X, W1, b1, bufA, N);
  k_smooth_init<<<2048, 256, 0, stream>>>(bufA, dis, bufB, N, Chid);
  k_edge_scatter<<<4096, 256, 0, stream>>>(bufA, dis, src, dst, bufB, E, Chid);
  k_relu<<<(N * Chid + 255) / 256, 256, 0, stream>>>(bufB, out1, N * Chid);

  // layer 2
  k_gemm2<<<(N + 31) / 32, 192, 0, stream>>>(out1, W2, b2, bufA, N);
  k_smooth_init<<<2048, 256, 0, stream>>>(bufA, dis, out2, N, Cout);
  k_edge_scatter<<<4096, 256, 0, stream>>>(bufA, dis, src, dst, out2, E, Cout);
}